// GAT_81999515615278
// MI455X (gfx1250) — compile-verified
//
#include <hip/hip_runtime.h>
#include <hip/hip_bf16.h>

typedef __attribute__((ext_vector_type(2))) float v2f;
typedef __attribute__((ext_vector_type(8))) float v8f;

#define HID 128
#define HEADS 4
#define HEAD_DIM 32
#define NEG_SLOPE 0.2f
#define BN_EPS 1e-5f
#define LDS_STRIDE 132   // 128 + 4 pad: breaks 64-bank periodicity of 128-float rows

// ---------------- float atomic max via int ordering trick ----------------
__device__ __forceinline__ void atomicMaxF(float* addr, float val) {
    if (val >= 0.0f)
        atomicMax((int*)addr, __float_as_int(val));
    else
        atomicMin((unsigned int*)addr, __float_as_uint(val));
}

// ---------------- fused GEMM: C[nrows x ncol] = act(A[nrows x 128] @ W[128 x ncol] + bias)
// block = 256 threads (8 wave32s); each block covers 128 rows.
// wave w computes rows [blk*128 + 16w, +16) for all col tiles using
// V_WMMA_F32_16X16X4_F32 (fp32 matrix cores), K=128 -> 32 WMMAs per tile.
// All A and B fragments for a tile are preloaded into VGPRs so the LDS loads
// pipeline (dscnt drains gradually) and the 32-WMMA accumulation chain issues
// back-to-back instead of stalling on s_wait_dscnt 0 per WMMA.
__global__ void gat_gemm128(const float* __restrict__ A, const float* __restrict__ W,
                            const float* __restrict__ bias, float* __restrict__ C,
                            int nrows, int ncol, int do_relu) {
    extern __shared__ float lds[];
    float* sA = lds;                       // 128 x LDS_STRIDE
    float* sW = lds + 128 * LDS_STRIDE;    // 128 x LDS_STRIDE
    const int tid  = threadIdx.x;
    const int row0 = blockIdx.x * 128;
    const int ct_n = (ncol + 15) >> 4;
    const int cpad = ct_n << 4;

    // warm the cache path for this block's A tile (lowers to global_prefetch_b8)
    {
        int pr = row0 + (tid >> 1);
        if (pr < nrows)
            __builtin_prefetch(&A[(size_t)pr * 128 + (tid & 1) * 64], 0, 3);
    }

    // stage A tile (coalesced; zero-fill out-of-range rows)
    for (int i = tid; i < 128 * 128; i += 256) {
        int r = i >> 7, c = i & 127;
        int gr = row0 + r;
        sA[r * LDS_STRIDE + c] = (gr < nrows) ? A[(size_t)gr * 128 + c] : 0.0f;
    }
    // stage W (zero-pad cols up to col-tile boundary)
    for (int i = tid; i < 128 * cpad; i += 256) {
        int k = i / cpad, c = i - k * cpad;
        sW[k * LDS_STRIDE + c] = (c < ncol) ? W[(size_t)k * ncol + c] : 0.0f;
    }
    __syncthreads();

    const int wave = tid >> 5;
    const int lane = tid & 31;
    const int half = lane >> 4;     // 0: K pair {0,1}, 1: K pair {2,3}
    const int l16  = lane & 15;
    const int kb2  = half * 2;
    const int mrow = wave * 16 + l16;   // A row (both halves read same 16 rows)

    // Preload all 32 A fragments (ISA layout: lane m holds A[m][kb2], A[m][kb2+1])
    v2f afrag[32];
#pragma unroll
    for (int kt = 0; kt < 32; ++kt)
        afrag[kt] = *(const v2f*)&sA[mrow * LDS_STRIDE + kt * 4 + kb2];

    for (int ct = 0; ct < ct_n; ++ct) {
        const int n = ct * 16 + l16;    // B/C column for this lane

        // Preload all 32 B fragments (B layout: lane n holds B[kb2][n], B[kb2+1][n]).
        // Keeping these in registers decouples LDS latency from the WMMA chain.
        v2f bfrag[32];
#pragma unroll
        for (int kt = 0; kt < 32; ++kt) {
            const int k0 = kt * 4 + kb2;
            v2f b;
            b.x = sW[k0 * LDS_STRIDE + n];
            b.y = sW[(k0 + 1) * LDS_STRIDE + n];
            bfrag[kt] = b;
        }

        v8f acc = {};
#pragma unroll
        for (int kt = 0; kt < 32; ++kt) {
            acc = __builtin_amdgcn_wmma_f32_16x16x4_f32(
                false, afrag[kt], false, bfrag[kt], (short)0, acc, false, false);
        }

        // epilogue: C/D layout -> VGPR v holds M = 8*half + v
        const float bv = (bias != nullptr && n < ncol) ? bias[n] : 0.0f;
#pragma unroll
        for (int v = 0; v < 8; ++v) {
            int grow = row0 + wave * 16 + half * 8 + v;
            if (grow < nrows && n < ncol) {
                float x = acc[v] + bv;
                if (do_relu) x = x > 0.0f ? x : 0.0f;
                C[(size_t)grow * ncol + n] = x;
            }
        }
    }
}

// ---------------- per-(node,head) attention scores: el = <hf, a_l>, er = <hf, a_r>
__global__ void gat_attn_scores(const float* __restrict__ hf,
                                const float* __restrict__ al,
                                const float* __restrict__ ar,
                                float* __restrict__ el, float* __restrict__ er,
                                int nnodes) {
    int idx = blockIdx.x * blockDim.x + threadIdx.x;
    if (idx >= nnodes * HEADS) return;
    int node = idx >> 2, head = idx & 3;
    const float* hp  = hf + (size_t)node * HID + head * HEAD_DIM;
    const float* alp = al + head * HEAD_DIM;
    const float* arp = ar + head * HEAD_DIM;
    float sl = 0.0f, sr = 0.0f;
#pragma unroll
    for (int d = 0; d < HEAD_DIM; ++d) {
        float x = hp[d];
        sl += x * alp[d];
        sr += x * arp[d];
    }
    el[idx] = sl;
    er[idx] = sr;
}

__global__ void gat_fill(float* __restrict__ p, float v, long long n) {
    long long i = (long long)blockIdx.x * blockDim.x + threadIdx.x;
    if (i < n) p[i] = v;
}

// ---------------- edge phase 1: e = leaky_relu(el[src]+er[dst]); segment max over dst
__global__ void gat_edge_max(const int* __restrict__ src, const int* __restrict__ dst,
                             const float* __restrict__ el, const float* __restrict__ er,
                             float* __restrict__ ebuf, float* __restrict__ maxb, int nedges) {
    int idx = blockIdx.x * blockDim.x + threadIdx.x;
    if (idx >= nedges * HEADS) return;
    int e = idx >> 2, h = idx & 3;
    int s = src[e], d = dst[e];
    float x = el[s * HEADS + h] + er[d * HEADS + h];
    x = x > 0.0f ? x : NEG_SLOPE * x;
    ebuf[idx] = x;
    atomicMaxF(&maxb[d * HEADS + h], x);
}

// ---------------- edge phase 2: ex = exp(e - m[dst]); segment sum over dst
__global__ void gat_edge_expsum(const int* __restrict__ dst, float* __restrict__ ebuf,
                                const float* __restrict__ maxb, float* __restrict__ sumb,
                                int nedges) {
    int idx = blockIdx.x * blockDim.x + threadIdx.x;
    if (idx >= nedges * HEADS) return;
    int e = idx >> 2, h = idx & 3;
    int d = dst[e];
    float ex = __expf(ebuf[idx] - maxb[d * HEADS + h]);
    ebuf[idx] = ex;
    atomicAdd(&sumb[d * HEADS + h], ex);
}

// ---------------- edge phase 3: agg[dst] += hf[src] * alpha  (128 threads per edge)
__global__ void gat_edge_aggregate(const int* __restrict__ src, const int* __restrict__ dst,
                                   const float* __restrict__ hf, const float* __restrict__ ebuf,
                                   const float* __restrict__ sumb, float* __restrict__ agg,
                                   int nedges) {
    int t = blockIdx.x * blockDim.x + threadIdx.x;
    int e = t >> 7;
    if (e >= nedges) return;
    int j = t & 127, h = j >> 5;
    int s = src[e], d = dst[e];
    float alpha = ebuf[e * HEADS + h] / sumb[d * HEADS + h];
    atomicAdd(&agg[(size_t)d * HID + j], hf[(size_t)s * HID + j] * alpha);
}

// ---------------- h_pre = agg + bias (+ residual); stored as new h_last in R
__global__ void gat_bias_res(const float* __restrict__ agg, const float* __restrict__ bias,
                             float* __restrict__ R, int n, int use_res) {
    int idx = blockIdx.x * blockDim.x + threadIdx.x;
    if (idx >= n) return;
    int c = idx & 127;
    float v = agg[idx] + bias[c];
    if (use_res) v += R[idx];
    R[idx] = v;
}

// ---------------- BN stats: 128 threads = 128 columns, coalesced row sweep
__global__ void gat_bn_stats(const float* __restrict__ R, float* __restrict__ stat, int nnodes) {
    int c = threadIdx.x;   // 0..127
    float s = 0.0f, q = 0.0f;
    for (int r = blockIdx.x; r < nnodes; r += gridDim.x) {
        float x = R[(size_t)r * HID + c];
        s += x;
        q += x * x;
    }
    atomicAdd(&stat[c], s);
    atomicAdd(&stat[HID + c], q);
}

// ---------------- h = relu(gamma * (x - mean) * rsqrt(var + eps) + beta)
__global__ void gat_bn_apply(const float* __restrict__ R, const float* __restrict__ stat,
                             const float* __restrict__ gamma, const float* __restrict__ beta,
                             float* __restrict__ A, int n, float invN) {
    int idx = blockIdx.x * blockDim.x + threadIdx.x;
    if (idx >= n) return;
    int c = idx & 127;
    float mean = stat[c] * invN;
    float var  = stat[HID + c] * invN - mean * mean;
    float v = gamma[c] * (R[idx] - mean) * rsqrtf(var + BN_EPS) + beta[c];
    A[idx] = v > 0.0f ? v : 0.0f;
}

extern "C" void kernel_launch(void* const* d_in, const int* in_sizes, int n_in,
                              void* d_out, int out_size, void* d_ws, size_t ws_size,
                              hipStream_t stream) {
    const float* feat   = (const float*)d_in[0];
    const int*   src    = (const int*)d_in[1];
    const int*   dst    = (const int*)d_in[2];
    const float* W_enc  = (const float*)d_in[3];
    const float* b_enc  = (const float*)d_in[4];
    const float* Ws     = (const float*)d_in[5];
    const float* a_ls   = (const float*)d_in[6];
    const float* a_rs   = (const float*)d_in[7];
    const float* biases = (const float*)d_in[8];
    const float* gammas = (const float*)d_in[9];
    const float* betas  = (const float*)d_in[10];
    const float* W_pred = (const float*)d_in[11];
    const float* b_pred = (const float*)d_in[12];

    const int N    = in_sizes[0] / HID;   // F_IN == HID == 128
    const int E    = in_sizes[1];
    const int NCLS = in_sizes[12];

    // workspace layout (fp32): h | hf | h_last | el | er | max | sum | ebuf | stats
    float* A  = (float*)d_ws;
    float* B  = A  + (size_t)N * HID;
    float* R  = B  + (size_t)N * HID;
    float* el = R  + (size_t)N * HID;
    float* er = el + (size_t)N * HEADS;
    float* mx = er + (size_t)N * HEADS;
    float* sm = mx + (size_t)N * HEADS;
    float* eb = sm + (size_t)N * HEADS;
    float* st = eb + (size_t)E * HEADS;

    const int gemmLds = 2 * 128 * LDS_STRIDE * (int)sizeof(float);
    const int gblocks = (N + 127) / 128;
    const int nh  = N * HID;
    const int na  = N * HEADS;
    const int ne4 = E * HEADS;

    // encoder: h = relu(feat @ W_enc + b_enc)
    gat_gemm128<<<gblocks, 256, gemmLds, stream>>>(feat, W_enc, b_enc, A, N, HID, 1);

    for (int L = 0; L < 3; ++L) {
        // hf = h @ W  (into B); h in A is dead afterwards -> reuse A as agg buffer
        gat_gemm128<<<gblocks, 256, gemmLds, stream>>>(
            A, Ws + (size_t)L * HID * HID, nullptr, B, N, HID, 0);

        gat_attn_scores<<<(na + 255) / 256, 256, 0, stream>>>(
            B, a_ls + L * HEADS * HEAD_DIM, a_rs + L * HEADS * HEAD_DIM, el, er, N);

        gat_fill<<<(na + 255) / 256, 256, 0, stream>>>(mx, -__builtin_huge_valf(), na);
        gat_fill<<<(na + 255) / 256, 256, 0, stream>>>(sm, 0.0f, na);
        gat_fill<<<(nh + 255) / 256, 256, 0, stream>>>(A, 0.0f, (long long)nh);

        gat_edge_max<<<(ne4 + 255) / 256, 256, 0, stream>>>(src, dst, el, er, eb, mx, E);
        gat_edge_expsum<<<(ne4 + 255) / 256, 256, 0, stream>>>(dst, eb, mx, sm, E);

        long long tagg = (long long)E * HID;
        gat_edge_aggregate<<<(unsigned)((tagg + 255) / 256), 256, 0, stream>>>(
            src, dst, B, eb, sm, A, E);

        // R = agg + bias (+ previous h_last for L>0); R becomes new h_last
        gat_bias_res<<<(nh + 255) / 256, 256, 0, stream>>>(
            A, biases + L * HID, R, nh, L > 0 ? 1 : 0);

        gat_fill<<<1, 256, 0, stream>>>(st, 0.0f, 2 * HID);
        gat_bn_stats<<<512, 128, 0, stream>>>(R, st, N);
        gat_bn_apply<<<(nh + 255) / 256, 256, 0, stream>>>(
            R, st, gammas + L * HID, betas + L * HID, A, nh, 1.0f / (float)N);
    }

    // predictor: out = h @ W_pred + b_pred  ([128 x 40], padded to 3 col tiles)
    gat_gemm128<<<gblocks, 256, gemmLds, stream>>>(A, W_pred, b_pred, (float*)d_out, N, NCLS, 0);
}